// SpectralFeature_29429115912880
// MI455X (gfx1250) — compile-verified
//
#include <hip/hip_runtime.h>
#include <math.h>

typedef __attribute__((ext_vector_type(2))) float v2f;
typedef __attribute__((ext_vector_type(8))) float v8f;

#define N_FFT      512
#define HOP        128
#define LEN        160000
#define NB         16
#define NC         4
#define NT         1251          // frames
#define NF         257           // rfft bins
#define FPAD       272           // 17 * 16
#define OUTW       (4*NC*NF)     // 4112
#define TT_TILES   79            // ceil(1251/16)
#define FT_TILES   17
#define LDS_STRIDE 516           // 512 + 4 dwords pad -> conflict-free A frags
#define PI_F       3.14159265358979323846f

__device__ __forceinline__ int reflect_idx(int i) {
    int j = i - N_FFT / 2;
    j = (j < 0) ? -j : j;
    j = (j >= LEN) ? (2 * LEN - 2 - j) : j;
    return j;
}

// ws layout (floats):
//   [BreP 512*272]  packed pairs: BreP[(j*FPAD + f)*2 + p] = Bre[2j+p][f]
//   [BimP 512*272]  same packing
//   [svre 4*257][svim 4*257]
// Packing means one global_load_b64 per B fragment per k-step in the GEMM.

__global__ void build_basis(float* __restrict__ BreP, float* __restrict__ BimP) {
    int idx = blockIdx.x * blockDim.x + threadIdx.x;
    if (idx >= N_FFT * FPAD) return;
    int k = idx / FPAD, f = idx % FPAD;
    float re = 0.f, im = 0.f;
    if (f < NF) {
        float w  = 0.5f * (1.0f - __cosf(2.0f * PI_F * (float)k / (float)N_FFT));
        int   m  = (k * f) & (N_FFT - 1);              // exact twiddle reduction
        float th = 2.0f * PI_F * (float)m / (float)N_FFT;
        re =  w * __cosf(th);
        im = -w * __sinf(th);
    }
    int j = k >> 1, p = k & 1;
    size_t o = ((size_t)j * FPAD + f) * 2 + p;         // fragment-ordered layout
    BreP[o] = re;
    BimP[o] = im;
}

__global__ void build_sv(const float* __restrict__ angle, const float* __restrict__ mic,
                         float* __restrict__ svre, float* __restrict__ svim) {
    __shared__ float td[NC];
    int tid = threadIdx.x;
    if (tid < NC) {
        float half_pi = 1.57079632679489662f;
        float s90 = __sinf(half_pi);
        float c90 = __cosf(half_pi);   // matches jnp.cos(pi/2) in f32 (tiny, nonzero)
        float sum = 0.f;
        for (int b = 0; b < NB; ++b) {
            float rad  = -angle[b] * (PI_F / 180.0f);
            float loc0 = __cosf(rad) * s90;
            float loc1 = __sinf(rad) * s90;
            float loc2 = 1.0f * c90;   // DIST * cos90
            float d0 = mic[(b * NC + tid) * 3 + 0] - loc0;
            float d1 = mic[(b * NC + tid) * 3 + 1] - loc1;
            float d2 = mic[(b * NC + tid) * 3 + 2] - loc2;
            sum += d0 * d0 + d1 * d1 + d2 * d2;
        }
        td[tid] = sqrtf(sum);          // reference sums over batch axis too
    }
    __syncthreads();
    float t0   = td[0];
    float coef = -2.0f * PI_F * 16000.0f / (512.0f * 340.4f);
    for (int i = tid; i < NC * NF; i += blockDim.x) {
        int c = i / NF, f = i % NF;
        float ph = coef * (float)f * (td[c] - t0);
        svre[i] = __cosf(ph) * 0.5f;   // / ||sv||_C  (= 2 exactly)
        svim[i] = __sinf(ph) * 0.5f;
    }
}

__global__ __launch_bounds__(544)
void stft_wmma(const float* __restrict__ x,
               const float* __restrict__ BreP, const float* __restrict__ BimP,
               const float* __restrict__ svre, const float* __restrict__ svim,
               float* __restrict__ out) {
    __shared__ float As[16 * LDS_STRIDE];   // 33 KB of 320 KB

    const int tt  = blockIdx.x;             // frame tile
    const int bc  = blockIdx.y;             // batch*channel
    const int b   = bc >> 2;
    const int c   = bc & 3;
    const int tid = threadIdx.x;

    const float* xrow = x + (size_t)bc * LEN;

    // Stage the 16x512 f32 A tile (reflect-padded; window folded into B).
    for (int idx = tid; idx < 16 * N_FFT; idx += 544) {
        int m = idx >> 9;
        int k = idx & (N_FFT - 1);
        int t = tt * 16 + m;
        float v = 0.f;
        if (t < NT) v = xrow[reflect_idx(t * HOP + k)];
        As[m * LDS_STRIDE + k] = v;
    }
    __syncthreads();

    const int wave = tid >> 5;              // 0..16  -> frequency tile
    const int lane = tid & 31;
    const int half = lane >> 4;             // K-half selector
    const int l16  = lane & 15;             // M (A/D) or N (B/D) index
    const int ft   = wave;
    const int fcol = ft * 16 + l16;

    v8f cRe = {};
    v8f cIm = {};
    const float* arow = &As[l16 * LDS_STRIDE];
    const v2f* __restrict__ brP = (const v2f*)BreP;   // [j*FPAD + f] pairs
    const v2f* __restrict__ biP = (const v2f*)BimP;

    for (int k0 = 0; k0 < N_FFT; k0 += 4) {
        int ka   = k0 + half * 2;                      // = 2 * jidx
        int jidx = (k0 >> 1) + half;
        v2f a  = *(const v2f*)(arow + ka);             // ds_load_b64
        v2f br = brP[(size_t)jidx * FPAD + fcol];      // global_load_b64
        v2f bi = biP[(size_t)jidx * FPAD + fcol];      // global_load_b64
        cRe = __builtin_amdgcn_wmma_f32_16x16x4_f32(false, a, false, br, (short)0, cRe, false, false);
        cIm = __builtin_amdgcn_wmma_f32_16x16x4_f32(false, a, false, bi, (short)0, cIm, false, false);
    }

    if (fcol < NF) {
        float svr = svre[c * NF + fcol];
        float svi = svim[c * NF + fcol];
#pragma unroll
        for (int r = 0; r < 8; ++r) {
            int t = tt * 16 + half * 8 + r;   // D: VGPR r -> M = 8*half + r
            if (t < NT) {
                size_t base = ((size_t)b * NT + t) * OUTW;
                // Output is streamed once and never re-read: NT stores keep
                // x (41 MB) and the basis resident in the 192 MB L2.
                __builtin_nontemporal_store(cRe[r], &out[base + (size_t)(0 * NC + c) * NF + fcol]);
                __builtin_nontemporal_store(cIm[r], &out[base + (size_t)(1 * NC + c) * NF + fcol]);
                __builtin_nontemporal_store(svr,    &out[base + (size_t)(2 * NC + c) * NF + fcol]);
                __builtin_nontemporal_store(svi,    &out[base + (size_t)(3 * NC + c) * NF + fcol]);
            }
        }
    }
}

extern "C" void kernel_launch(void* const* d_in, const int* in_sizes, int n_in,
                              void* d_out, int out_size, void* d_ws, size_t ws_size,
                              hipStream_t stream) {
    const float* x     = (const float*)d_in[0];
    const float* angle = (const float*)d_in[1];
    const float* mic   = (const float*)d_in[2];
    float* out = (float*)d_out;
    float* ws  = (float*)d_ws;

    float* BreP = ws;
    float* BimP = BreP + N_FFT * FPAD;
    float* svre = BimP + N_FFT * FPAD;
    float* svim = svre + NC * NF;

    build_basis<<<(N_FFT * FPAD + 255) / 256, 256, 0, stream>>>(BreP, BimP);
    build_sv<<<1, 256, 0, stream>>>(angle, mic, svre, svim);

    dim3 grid(TT_TILES, NB * NC);
    stft_wmma<<<grid, 544, 0, stream>>>(x, BreP, BimP, svre, svim, out);
}